// FBGAT_Layer_721554506157
// MI455X (gfx1250) — compile-verified
//
#include <hip/hip_runtime.h>
#include <hip/hip_bf16.h>

// ---------------------------------------------------------------------------
// FBGAT layer for MI455X (gfx1250, wave32, WMMA, async global->LDS).
//
// Pipeline (reassociated: ((Di·L)·Di)·R == (Di·L)·(Di·R), saves one 4096^3 GEMM):
//   T  = d_inv @ lap                  [4096x4096]  (WMMA f32 GEMM, dominant)
//   R  = relu(x @ W_high^T)           [4096x256 ]  (WMMA f32 GEMM, B^T, relu)
//   P  = d_inv @ R                    [4096x256 ]  (WMMA f32 GEMM)
//   Hh = T @ P                        [4096x256 ]  (WMMA f32 GEMM)
//   h  = x @ W_gat^T                  [4096x256 ]  (WMMA f32 GEMM, B^T)
//   a_src/a_dst per (node,head)       (LDS reduction)
//   edge scores + segment max (atomicMax on monotone key)
//   exp + segment sum (atomicAdd f32)
//   weighted scatter into Hl (coalesced atomicAdd f32)
//   out = aL*(Hl + bias) + aH*Hh
//
// GEMM: 256 thr = 8 wave32, block tile 128x128, wave tile 32x64 (2x4 frags),
// K-slab 16, LDS ping-pong double buffer filled by GLOBAL_LOAD_ASYNC_TO_LDS
// (ASYNCcnt; no VGPR staging, no spills). Both A and B stored [row][k]
// stride-20 in LDS so every fragment is a conflict-free ds_load_b64.
// ---------------------------------------------------------------------------

typedef __attribute__((ext_vector_type(2))) float v2f;
typedef __attribute__((ext_vector_type(8))) float v8f;

#define GN    4096      // nodes
#define GIN   256       // input dim
#define GHC   256       // H*C
#define GH    4         // heads
#define BM    128
#define BN    128
#define BK    16
#define LSTR  20        // LDS stride (floats): 16B-aligned rows, conflict-free b64 reads

#define WMMA_F32(a, b, c) \
    __builtin_amdgcn_wmma_f32_16x16x4_f32(false, (a), false, (b), (short)0, (c), false, false)

__device__ __forceinline__ void async_ld_b128(unsigned lds_off, unsigned long long gaddr) {
    asm volatile("global_load_async_to_lds_b128 %0, %1, off"
                 :: "v"(lds_off), "v"(gaddr) : "memory");
}
__device__ __forceinline__ void async_ld_b32(unsigned lds_off, unsigned long long gaddr) {
    asm volatile("global_load_async_to_lds_b32 %0, %1, off"
                 :: "v"(lds_off), "v"(gaddr) : "memory");
}
__device__ __forceinline__ void wait_async0() {
    asm volatile("s_wait_asynccnt 0" ::: "memory");
}
__device__ __forceinline__ unsigned lds_off32(const void* p) {   // flat->LDS offset
    return (unsigned)(size_t)p;
}

// ------------------------- WMMA FP32 GEMM ----------------------------------
// C[M,N] = op(A[M,K] @ B), A row-major (lda=K).
// OPB==0: B is [K,N] row-major (ldb=N).  OPB==1: B is [N,K] row-major (ldb=K) => A@B^T.
// Requires M%128==0, N%128==0, K%16==0 (true for all uses here).
template<int OPB, int DO_RELU>
__global__ __launch_bounds__(256, 1) void gemm_wmma_kernel(
    const float* __restrict__ A, const float* __restrict__ B, float* __restrict__ C,
    int K, int lda, int ldb, int ldc)
{
    __shared__ float As[2][BM * LSTR];      // [m][k], stride 20
    __shared__ float Bs[2][BN * LSTR];      // [n][k], stride 20 (B transposed in LDS)

    const int tid  = threadIdx.x;
    const int lane = tid & 31;
    const int wave = tid >> 5;             // 0..7
    const int wm   = (wave >> 1) * 32;     // wave M offset: 0,32,64,96
    const int wn   = (wave & 1) * 64;      // wave N offset: 0,64
    const int half = lane >> 4;            // 0/1
    const int lm   = lane & 15;

    const int rowBlk = blockIdx.y * BM;
    const int colBlk = blockIdx.x * BN;

    // A: 128x16 slab, 2 x b128 per thread, coalesced 16B rows
    const int aFlat0 = tid * 4;
    const int aFlat1 = (256 + tid) * 4;
    const int ar0 = aFlat0 >> 4, ac0 = aFlat0 & (BK - 1);
    const int ar1 = aFlat1 >> 4, ac1 = aFlat1 & (BK - 1);
    // B(opB=0): one n-column of the 16x128 slab per thread: 8 coalesced async b32
    const int bn0 = tid & (BN - 1);
    const int bkh = (tid >> 7) * 8;        // 0 or 8
    // B(opB=1): thread covers fixed n, 8 consecutive k: 2 x async b128
    const int tn1 = tid >> 1;
    const int tk1 = (tid & 1) * 8;

    v8f acc[2][4] = {};

    // async DMA fill of LDS buffer `buf` with K-slab starting at k0
    auto fillAsync = [&](int buf, int k0) {
        async_ld_b128(lds_off32(&As[buf][ar0 * LSTR + ac0]),
                      (unsigned long long)(size_t)(A + (size_t)(rowBlk + ar0) * lda + k0 + ac0));
        async_ld_b128(lds_off32(&As[buf][ar1 * LSTR + ac1]),
                      (unsigned long long)(size_t)(A + (size_t)(rowBlk + ar1) * lda + k0 + ac1));
        if (OPB == 0) {
            #pragma unroll
            for (int j = 0; j < 8; ++j)
                async_ld_b32(lds_off32(&Bs[buf][bn0 * LSTR + bkh + j]),
                             (unsigned long long)(size_t)(B + (size_t)(k0 + bkh + j) * ldb + colBlk + bn0));
        } else {
            async_ld_b128(lds_off32(&Bs[buf][tn1 * LSTR + tk1]),
                          (unsigned long long)(size_t)(B + (size_t)(colBlk + tn1) * ldb + k0 + tk1));
            async_ld_b128(lds_off32(&Bs[buf][tn1 * LSTR + tk1 + 4]),
                          (unsigned long long)(size_t)(B + (size_t)(colBlk + tn1) * ldb + k0 + tk1 + 4));
        }
    };

    // ---- prologue: DMA slab 0 into buffer 0 ----
    fillAsync(0, 0);
    wait_async0();
    __syncthreads();

    int cur = 0;
    for (int k0 = 0; k0 < K; k0 += BK) {
        const bool more = (k0 + BK) < K;
        if (more) fillAsync(cur ^ 1, k0 + BK);   // DMA next slab while computing

        const float* as = &As[cur][0];
        const float* bs = &Bs[cur][0];
        #pragma unroll
        for (int kk = 0; kk < BK; kk += 4) {
            // A frags: lane = row (wm+i*16+lm), k = kk + half*2 + {0,1}
            v2f a0 = *reinterpret_cast<const v2f*>(&as[(wm +  0 + lm) * LSTR + kk + half * 2]);
            v2f a1 = *reinterpret_cast<const v2f*>(&as[(wm + 16 + lm) * LSTR + kk + half * 2]);
            // B frags: lane = col (wn+j*16+lm), k = kk + half*2 + {0,1}
            v2f b0 = *reinterpret_cast<const v2f*>(&bs[(wn +  0 + lm) * LSTR + kk + half * 2]);
            v2f b1 = *reinterpret_cast<const v2f*>(&bs[(wn + 16 + lm) * LSTR + kk + half * 2]);
            v2f b2 = *reinterpret_cast<const v2f*>(&bs[(wn + 32 + lm) * LSTR + kk + half * 2]);
            v2f b3 = *reinterpret_cast<const v2f*>(&bs[(wn + 48 + lm) * LSTR + kk + half * 2]);
            acc[0][0] = WMMA_F32(a0, b0, acc[0][0]);
            acc[0][1] = WMMA_F32(a0, b1, acc[0][1]);
            acc[0][2] = WMMA_F32(a0, b2, acc[0][2]);
            acc[0][3] = WMMA_F32(a0, b3, acc[0][3]);
            acc[1][0] = WMMA_F32(a1, b0, acc[1][0]);
            acc[1][1] = WMMA_F32(a1, b1, acc[1][1]);
            acc[1][2] = WMMA_F32(a1, b2, acc[1][2]);
            acc[1][3] = WMMA_F32(a1, b3, acc[1][3]);
        }

        if (more) wait_async0();                 // my fills of buf^1 complete
        __syncthreads();                         // everyone done reading cur / filling cur^1
        cur ^= 1;
    }

    // ---- store: C/D layout => row = r + 8*half, col = lm --------------------
    #pragma unroll
    for (int i = 0; i < 2; ++i) {
        #pragma unroll
        for (int j = 0; j < 4; ++j) {
            const int col = colBlk + wn + j * 16 + lm;
            #pragma unroll
            for (int r = 0; r < 8; ++r) {
                const int row = rowBlk + wm + i * 16 + half * 8 + r;
                float v = acc[i][j][r];
                if (DO_RELU) v = v > 0.0f ? v : 0.0f;
                C[(size_t)row * ldc + col] = v;
            }
        }
    }
}

// ------------------------- small kernels -----------------------------------

__global__ void zero_kernel(float* __restrict__ p, int n) {
    int g = blockIdx.x * blockDim.x + threadIdx.x;
    if (g < n) p[g] = 0.0f;
}

// a_src[n][h] = sum_c h[n][h][c]*att_src[h][c] ; same for a_dst. 1 block / node.
__global__ __launch_bounds__(256) void attn_coeff_kernel(
    const float* __restrict__ h, const float* __restrict__ att_s,
    const float* __restrict__ att_d, float* __restrict__ a_s, float* __restrict__ a_d)
{
    __shared__ float ss[256];
    __shared__ float sd[256];
    const int n = blockIdx.x, t = threadIdx.x;
    const float v = h[(size_t)n * GHC + t];
    ss[t] = v * att_s[t];
    sd[t] = v * att_d[t];
    __syncthreads();
    #pragma unroll
    for (int s = 32; s > 0; s >>= 1) {      // reduce each 64-wide head chunk
        if ((t & 63) < s) { ss[t] += ss[t + s]; sd[t] += sd[t + s]; }
        __syncthreads();
    }
    if ((t & 63) == 0) {
        const int hd = t >> 6;
        a_s[n * GH + hd] = ss[t];
        a_d[n * GH + hd] = sd[t];
    }
}

__device__ __forceinline__ unsigned fkey(float f) {   // monotone float->uint key
    unsigned b = __float_as_uint(f);
    return (b & 0x80000000u) ? ~b : (b | 0x80000000u);
}
__device__ __forceinline__ float funkey(unsigned k) {
    unsigned b = (k & 0x80000000u) ? (k & 0x7FFFFFFFu) : ~k;
    return __uint_as_float(b);
}

// e[i][h] = leaky_relu(a_src[src_i][h] + a_dst[dst_i][h]); segment max via atomicMax
__global__ void edge_score_kernel(
    const int* __restrict__ src, const int* __restrict__ dst,
    const float* __restrict__ a_s, const float* __restrict__ a_d,
    float* __restrict__ e, unsigned* __restrict__ mkey, int E, int EN)
{
    int g = blockIdx.x * blockDim.x + threadIdx.x;
    if (g >= EN * GH) return;
    const int i = g >> 2, hd = g & 3;
    const int s = (i < E) ? src[i] : (i - E);     // self loops appended
    const int d = (i < E) ? dst[i] : (i - E);
    float v = a_s[s * GH + hd] + a_d[d * GH + hd];
    v = v > 0.0f ? v : 0.2f * v;                  // leaky_relu(0.2)
    e[g] = v;
    atomicMax(&mkey[d * GH + hd], fkey(v));
}

// p = exp(e - m[dst]); denom[dst] += p  (p overwrites e)
__global__ void edge_exp_kernel(
    const int* __restrict__ dst, float* __restrict__ e,
    const unsigned* __restrict__ mkey, float* __restrict__ denom, int E, int EN)
{
    int g = blockIdx.x * blockDim.x + threadIdx.x;
    if (g >= EN * GH) return;
    const int i = g >> 2, hd = g & 3;
    const int d = (i < E) ? dst[i] : (i - E);
    const float m = funkey(mkey[d * GH + hd]);
    const float p = __expf(e[g] - m);
    e[g] = p;
    atomicAdd(&denom[d * GH + hd], p);
}

// Hl[dst] += alpha * h[src]; 1 block / edge, 256 threads (coalesced f32 atomics)
__global__ __launch_bounds__(256) void aggregate_kernel(
    const int* __restrict__ src, const int* __restrict__ dst,
    const float* __restrict__ p, const float* __restrict__ denom,
    const float* __restrict__ h, float* __restrict__ Hl, int E)
{
    const int i = blockIdx.x;
    const int t = threadIdx.x;
    const int hd = t >> 6;
    const int s = (i < E) ? src[i] : (i - E);
    const int d = (i < E) ? dst[i] : (i - E);
    const float alpha = p[i * GH + hd] / (denom[d * GH + hd] + 1e-16f);
    atomicAdd(&Hl[(size_t)d * GHC + t], alpha * h[(size_t)s * GHC + t]);
}

// out = aL*(Hl + bias) + aH*Hh
__global__ void combine_kernel(
    const float* __restrict__ Hl, const float* __restrict__ Hh,
    const float* __restrict__ bias, const float* __restrict__ aLp,
    const float* __restrict__ aHp, float* __restrict__ out, int n)
{
    int g = blockIdx.x * blockDim.x + threadIdx.x;
    if (g >= n) return;
    const float aL = aLp[0], aH = aHp[0];
    out[g] = aL * (Hl[g] + bias[g & (GHC - 1)]) + aH * Hh[g];
}

// ------------------------- host launcher -----------------------------------

extern "C" void kernel_launch(void* const* d_in, const int* in_sizes, int n_in,
                              void* d_out, int out_size, void* d_ws, size_t ws_size,
                              hipStream_t stream)
{
    const float* x       = (const float*)d_in[0];
    const int*   eidx    = (const int*)  d_in[1];
    const float* lap     = (const float*)d_in[2];
    const float* d_inv   = (const float*)d_in[3];
    const float* W_high  = (const float*)d_in[4];
    const float* W_gat   = (const float*)d_in[5];
    const float* att_src = (const float*)d_in[6];
    const float* att_dst = (const float*)d_in[7];
    const float* bias    = (const float*)d_in[8];
    const float* aL      = (const float*)d_in[9];
    const float* aH      = (const float*)d_in[10];

    const int E  = in_sizes[1] / 2;
    const int EN = E + GN;
    const int* src = eidx;
    const int* dst = eidx + E;

    // workspace layout (floats)
    float* W    = (float*)d_ws;
    float* T    = W;                               // 4096*4096
    float* R    = T   + (size_t)GN * GN;           // 4096*256
    float* P    = R   + (size_t)GN * GHC;
    float* Hh   = P   + (size_t)GN * GHC;
    float* hbuf = Hh  + (size_t)GN * GHC;
    float* Hl   = hbuf+ (size_t)GN * GHC;
    float* a_s  = Hl  + (size_t)GN * GHC;          // 4096*4
    float* a_d  = a_s + (size_t)GN * GH;
    float* ebuf = a_d + (size_t)GN * GH;           // EN*4
    unsigned* mkey = (unsigned*)(ebuf + (size_t)EN * GH);   // 4096*4
    float* denom   = (float*)(mkey + (size_t)GN * GH);      // 4096*4
    const size_t ws_need = (size_t)((denom + (size_t)GN * GH) - W) * sizeof(float);
    if (ws_size < ws_need) return;

    dim3 blk(256);

    // --- high-frequency path (one 4096^3 GEMM + three thin GEMMs) ---
    gemm_wmma_kernel<0,0><<<dim3(GN/BN,  GN/BM), blk, 0, stream>>>(d_inv, lap,    T,  GN,  GN,  GN,  GN);   // T = Di@L
    gemm_wmma_kernel<1,1><<<dim3(GHC/BN, GN/BM), blk, 0, stream>>>(x,     W_high, R,  GIN, GIN, GIN, GHC);  // R = relu(x@Wh^T)
    gemm_wmma_kernel<0,0><<<dim3(GHC/BN, GN/BM), blk, 0, stream>>>(d_inv, R,      P,  GN,  GN,  GHC, GHC);  // P = Di@R
    gemm_wmma_kernel<0,0><<<dim3(GHC/BN, GN/BM), blk, 0, stream>>>(T,     P,      Hh, GN,  GN,  GHC, GHC);  // Hh = T@P

    // --- GAT path ---
    gemm_wmma_kernel<1,0><<<dim3(GHC/BN, GN/BM), blk, 0, stream>>>(x,     W_gat,  hbuf, GIN, GIN, GIN, GHC); // h = x@Wg^T
    attn_coeff_kernel<<<GN, 256, 0, stream>>>(hbuf, att_src, att_dst, a_s, a_d);

    zero_kernel<<<(GN*GH + 255)/256, blk, 0, stream>>>((float*)mkey, GN*GH);
    zero_kernel<<<(GN*GH + 255)/256, blk, 0, stream>>>(denom, GN*GH);
    zero_kernel<<<(GN*GHC + 255)/256, blk, 0, stream>>>(Hl, GN*GHC);

    const int EG = EN * GH;
    edge_score_kernel<<<(EG + 255)/256, blk, 0, stream>>>(src, dst, a_s, a_d, ebuf, mkey, E, EN);
    edge_exp_kernel  <<<(EG + 255)/256, blk, 0, stream>>>(dst, ebuf, mkey, denom, E, EN);
    aggregate_kernel <<<EN, 256, 0, stream>>>(src, dst, ebuf, denom, hbuf, Hl, E);

    combine_kernel<<<(GN*GHC + 255)/256, blk, 0, stream>>>(Hl, Hh, bias, aL, aH, (float*)d_out, GN*GHC);
}